// QuantumConv_1090921693430
// MI455X (gfx1250) — compile-verified
//
#include <hip/hip_runtime.h>
#include <math.h>

typedef __attribute__((ext_vector_type(2))) float v2f;
typedef __attribute__((ext_vector_type(8))) float v8f;

#define NWIRE 6
#define NLAY  4
#define HPOOL 112
#define WPOOL 112
#define HFULL 224
#define WFULL 224
#define BATCH 8
#define OUTC  16
#define NPIX  (BATCH*HPOOL*WPOOL)   // 100352, divisible by 16

struct c32 { float x, y; };
__device__ __forceinline__ c32 cmul(c32 a, c32 b){ return {a.x*b.x - a.y*b.y, a.x*b.y + a.y*b.x}; }
__device__ __forceinline__ c32 cadd(c32 a, c32 b){ return {a.x+b.x, a.y+b.y}; }

// ---------------------------------------------------------------------------
// Kernel 1: simulate the fixed 6-qubit circuit on the 8 basis columns of the
// input subspace, fold phases + FC weights into T[16][64] (real GEMM matrix).
// One block, 64 threads; negligible cost, runs once per launch.
// ---------------------------------------------------------------------------
__global__ void qc_precompute(const float* __restrict__ wts,   // (4,6,3)
                              const float* __restrict__ fcw,   // (16,6)
                              float* __restrict__ T)           // (16,64)
{
    __shared__ c32  st[8][64];     // column k (input basis), amplitude i
    __shared__ float Ssh[6][64];   // S_w[k][k'] flattened k*8+k'
    const int t = threadIdx.x;     // 64 threads

    if (t < 8) {
        for (int i = 0; i < 64; ++i) st[t][i] = {0.f, 0.f};
        st[t][t * 8] = {1.f, 0.f};   // wires 3..5 = |0>, wires0..2 = bits of t (MSB first)
        for (int l = 0; l < NLAY; ++l) {
            // single-qubit rotations
            for (int w = 0; w < NWIRE; ++w) {
                const float phi = wts[(l*NWIRE + w)*3 + 0];
                const float th  = wts[(l*NWIRE + w)*3 + 1];
                const float om  = wts[(l*NWIRE + w)*3 + 2];
                const float ch = cosf(0.5f*th), sh = sinf(0.5f*th);
                const float ap = 0.5f*(phi+om), am = 0.5f*(phi-om);
                const c32 ep = {cosf(ap), -sinf(ap)};   // exp(-i(phi+om)/2)
                const c32 em = {cosf(am), -sinf(am)};   // exp(-i(phi-om)/2)
                const c32 u00 = { ep.x*ch,  ep.y*ch};
                const c32 u01 = {-em.x*sh,  em.y*sh};   // -conj(em)*s
                const c32 u10 = { em.x*sh,  em.y*sh};
                const c32 u11 = { ep.x*ch, -ep.y*ch};   // conj(ep)*c
                const int mask = 32 >> w;               // wire w is bit (5-w)
                for (int i = 0; i < 64; ++i) {
                    if (i & mask) continue;
                    const c32 s0 = st[t][i], s1 = st[t][i | mask];
                    st[t][i]        = cadd(cmul(u00, s0), cmul(u01, s1));
                    st[t][i | mask] = cadd(cmul(u10, s0), cmul(u11, s1));
                }
            }
            // CNOT ring, shift r
            const int r = l % (NWIRE - 1) + 1;
            for (int w = 0; w < NWIRE; ++w) {
                const int cm = 32 >> w;
                const int tm = 32 >> ((w + r) % NWIRE);
                for (int i = 0; i < 64; ++i) {
                    if ((i & cm) && !(i & tm)) {
                        const c32 tmp = st[t][i];
                        st[t][i] = st[t][i | tm];
                        st[t][i | tm] = tmp;
                    }
                }
            }
        }
    }
    __syncthreads();

    // S_w[k][k'] = Re[ i^(pc(k)-pc(k')) * sum_i sign_w(i) conj(A[i][k]) A[i][k'] ]
    {
        const int k  = t >> 3, k2 = t & 7;
        const int d  = (__popc(k) - __popc(k2)) & 3;
        for (int w = 0; w < NWIRE; ++w) {
            const int mask = 32 >> w;
            float ax = 0.f, ay = 0.f;
            for (int i = 0; i < 64; ++i) {
                const float sgn = (i & mask) ? -1.f : 1.f;
                const c32 a = st[k][i], b = st[k2][i];
                ax += sgn * (a.x*b.x + a.y*b.y);   // Re(conj(a)*b)
                ay += sgn * (a.x*b.y - a.y*b.x);   // Im(conj(a)*b)
            }
            float s;
            switch (d) { case 0: s = ax; break; case 1: s = -ay; break;
                         case 2: s = -ax; break; default: s = ay; }
            Ssh[w][t] = s;
        }
    }
    __syncthreads();

    // T[c][e] = sum_w fc_w[c][w] * S_w[e]
    for (int c = 0; c < OUTC; ++c) {
        float acc = 0.f;
        for (int w = 0; w < NWIRE; ++w) acc += fcw[c*NWIRE + w] * Ssh[w][t];
        T[c*64 + t] = acc;
    }
}

// ---------------------------------------------------------------------------
// Kernel 2: per-pixel pipeline. Each wave handles 16 pooled pixels.
// y[16ch][16pix] = T[16][64] @ rr[64][16pix] via 16 chained
// V_WMMA_F32_16X16X4_F32, then bias + LayerNorm (shfl_xor 16) + ReLU,
// write the 112x112x16 map to d_ws.
// ---------------------------------------------------------------------------
__global__ void qc_main(const float* __restrict__ x,
                        const float* __restrict__ T,
                        const float* __restrict__ fcb,
                        const float* __restrict__ lng,
                        const float* __restrict__ lnb,
                        float* __restrict__ yln)
{
    const int lane = threadIdx.x & 31;
    const int wave = blockIdx.x * (blockDim.x >> 5) + (threadIdx.x >> 5);
    const int p    = wave * 16 + (lane & 15);     // this lane's pixel

    const int b   = p / (HPOOL * WPOOL);
    const int rem = p - b * (HPOOL * WPOOL);
    const int hp  = rem / WPOOL;
    const int wp  = rem - hp * WPOOL;

    // 2x2 mean pool -> half-angles -> cos/sin per input channel
    float cs[3], sn[3];
    #pragma unroll
    for (int c = 0; c < 3; ++c) {
        const float* base = x + ((size_t)(b*3 + c)*HFULL + 2*hp) * WFULL + 2*wp;
        const float a = (base[0] + base[1] + base[WFULL] + base[WFULL+1]) * 0.25f;
        const float h = 0.5f * a;
        cs[c] = cosf(h);
        sn[c] = sinf(h);
    }
    // r_k = product over wire bits (wire0 = bit2 of k)
    float r[8];
    #pragma unroll
    for (int k = 0; k < 8; ++k)
        r[k] = ((k & 4) ? sn[0] : cs[0]) * ((k & 2) ? sn[1] : cs[1]) * ((k & 1) ? sn[2] : cs[2]);

    // WMMA f32 16x16x4 chain: lane supplies K = 4*kk + koff (+1)
    const int koff = (lane >= 16) ? 2 : 0;
    const int m    = lane & 15;                   // A-matrix row (output channel group row)
    v8f acc = {};
    #pragma unroll
    for (int kk = 0; kk < 16; ++kk) {
        const int k0 = 4*kk + koff;
        const v2f a = *(const v2f*)(T + m*64 + k0);       // A: T[m][k0], T[m][k0+1]
        v2f bv;
        bv[0] = r[k0 >> 3]       * r[k0 & 7];             // rr[k0]
        bv[1] = r[(k0+1) >> 3]   * r[(k0+1) & 7];         // rr[k0+1]
        acc = __builtin_amdgcn_wmma_f32_16x16x4_f32(
                  false, a, false, bv, (short)0, acc, false, false);
    }

    // Lane holds channels coff..coff+7 of its pixel
    const int coff = (lane >= 16) ? 8 : 0;
    float y[8];
    float s = 0.f, ss = 0.f;
    #pragma unroll
    for (int v = 0; v < 8; ++v) {
        y[v] = acc[v] + fcb[coff + v];
        s  += y[v];
        ss += y[v] * y[v];
    }
    s  += __shfl_xor(s,  16, 32);
    ss += __shfl_xor(ss, 16, 32);
    const float mu  = s * (1.f/16.f);
    const float var = ss * (1.f/16.f) - mu*mu;
    const float inv = rsqrtf(var + 1e-5f);

    float* outp = yln + (size_t)p * OUTC + coff;
    #pragma unroll
    for (int v = 0; v < 8; ++v) {
        const float yy = (y[v] - mu) * inv * lng[coff + v] + lnb[coff + v];
        outp[v] = fmaxf(yy, 0.f);
    }
}

// ---------------------------------------------------------------------------
// Kernel 3: 2x bilinear upsample (jax.image.resize semantics: triangle kernel
// with edge weight renormalization -> interior 0.25/0.75, edges copy).
// ---------------------------------------------------------------------------
__global__ void qc_resize(const float* __restrict__ yln, float* __restrict__ out)
{
    const int idx = blockIdx.x * blockDim.x + threadIdx.x;
    const int tot = BATCH * OUTC * HFULL * WFULL;
    if (idx >= tot) return;
    const int w = idx % WFULL;
    const int h = (idx / WFULL) % HFULL;
    const int c = (idx / (WFULL * HFULL)) % OUTC;
    const int b = idx / (WFULL * HFULL * OUTC);

    int h0, h1; float wh0, wh1;
    {
        const int hm = h >> 1;
        if ((h & 1) == 0) {
            if (hm == 0)      { h0 = 0;    h1 = 0;    wh0 = 1.f;   wh1 = 0.f;   }
            else              { h0 = hm-1; h1 = hm;   wh0 = 0.25f; wh1 = 0.75f; }
        } else {
            if (hm == HPOOL-1){ h0 = hm;   h1 = hm;   wh0 = 1.f;   wh1 = 0.f;   }
            else              { h0 = hm;   h1 = hm+1; wh0 = 0.75f; wh1 = 0.25f; }
        }
    }
    int w0, w1; float ww0, ww1;
    {
        const int wm = w >> 1;
        if ((w & 1) == 0) {
            if (wm == 0)      { w0 = 0;    w1 = 0;    ww0 = 1.f;   ww1 = 0.f;   }
            else              { w0 = wm-1; w1 = wm;   ww0 = 0.25f; ww1 = 0.75f; }
        } else {
            if (wm == WPOOL-1){ w0 = wm;   w1 = wm;   ww0 = 1.f;   ww1 = 0.f;   }
            else              { w0 = wm;   w1 = wm+1; ww0 = 0.75f; ww1 = 0.25f; }
        }
    }
    const size_t rowb = (size_t)b * HPOOL;
    const float v00 = yln[((rowb + h0)*WPOOL + w0)*OUTC + c];
    const float v01 = yln[((rowb + h0)*WPOOL + w1)*OUTC + c];
    const float v10 = yln[((rowb + h1)*WPOOL + w0)*OUTC + c];
    const float v11 = yln[((rowb + h1)*WPOOL + w1)*OUTC + c];
    out[idx] = wh0 * (ww0*v00 + ww1*v01) + wh1 * (ww0*v10 + ww1*v11);
}

// ---------------------------------------------------------------------------
extern "C" void kernel_launch(void* const* d_in, const int* in_sizes, int n_in,
                              void* d_out, int out_size, void* d_ws, size_t ws_size,
                              hipStream_t stream)
{
    const float* x   = (const float*)d_in[0];   // (8,3,224,224)
    const float* wts = (const float*)d_in[1];   // (4,6,3)
    const float* fcw = (const float*)d_in[2];   // (16,6)
    const float* fcb = (const float*)d_in[3];   // (16,)
    const float* lng = (const float*)d_in[4];   // (16,)
    const float* lnb = (const float*)d_in[5];   // (16,)
    float* out = (float*)d_out;                 // (8,16,224,224)

    float* T   = (float*)d_ws;                          // 16*64 floats = 4KB
    float* yln = (float*)((char*)d_ws + 4096);          // NPIX*16 floats ≈ 6.4MB

    qc_precompute<<<1, 64, 0, stream>>>(wts, fcw, T);

    // NPIX/16 = 6272 waves, 8 waves (256 threads) per block -> 784 blocks
    qc_main<<<NPIX / 16 / 8, 256, 0, stream>>>(x, T, fcb, lng, lnb, yln);

    const int tot = BATCH * OUTC * HFULL * WFULL;
    qc_resize<<<(tot + 255) / 256, 256, 0, stream>>>(yln, out);
}